// PhaseAdaptiveInput_55293408969356
// MI455X (gfx1250) — compile-verified
//
#include <hip/hip_runtime.h>

// Phase-adaptive NNUE-style sparse input layer for MI455X (gfx1250).
// Strategy: bucket-first gather (6x traffic cut), TDM gather-mode descriptors
// (TENSOR_LOAD_TO_LDS, 8 x 32-bit row indices per descriptor -> 3 TDM ops per
// sample) + S_WAIT_TENSORCNT, wave-per-sample wave32, readlane broadcasts.

typedef unsigned int u32;
typedef unsigned int u32x4 __attribute__((ext_vector_type(4)));
typedef unsigned int u32x8 __attribute__((ext_vector_type(8)));

#define ACTIVE 22
#define LPA 96
#define OUT_DIM 576
#define BUCKET_SIZE 10
#define WAVES_PER_BLOCK 8
#define ROW_BYTES (LPA * 4)   /* 384 */
#define ROWS_PER_DESC 8

__global__ __launch_bounds__(WAVES_PER_BLOCK * 32) void
nnue_phase_tdm_kernel(const int* __restrict__ cols,
                      const float* __restrict__ values,
                      const float* __restrict__ weight,
                      const float* __restrict__ bias,
                      const int* __restrict__ ply,
                      float* __restrict__ out,
                      int batch, u32 F)
{
    // 8 waves x 22 rows x 96 floats = 67.5 KB of the 320 KB WGP LDS
    __shared__ __align__(16) float smem[WAVES_PER_BLOCK * ACTIVE * LPA];

    const int lane = threadIdx.x & 31;
    const int wave = threadIdx.x >> 5;
    const int s    = blockIdx.x * WAVES_PER_BLOCK + wave;
    if (s >= batch) return;

    // bucket-first: only gather the 96 surviving columns of each row
    const int bucket = ply[s] / BUCKET_SIZE;       // wave-uniform
    const int cb     = bucket * LPA;

    // cooperative load of this sample's 22 (index, value) pairs
    int   myCol = 0;
    float myVal = 0.0f;
    if (lane < ACTIVE) {
        myCol = cols[s * ACTIVE + lane];
        myVal = values[s * ACTIVE + lane];
    }

    float* lds = smem + wave * (ACTIVE * LPA);

    // Wave-uniform descriptor ingredients (explicit readfirstlane so the "s"
    // asm constraints are fed genuinely uniform SGPR values).
    const unsigned long long gbase = (unsigned long long)(weight + cb);
    const u32 glo   = __builtin_amdgcn_readfirstlane((u32)gbase);
    const u32 ghi   = __builtin_amdgcn_readfirstlane((u32)(gbase >> 32));
    const u32 lbase = __builtin_amdgcn_readfirstlane((u32)(unsigned long long)lds);

    // 3 TDM gather-mode descriptors: 8 + 8 + 6 row indices of 384B each.
#pragma unroll
    for (int d = 0; d < 3; ++d) {
        const u32 nv = (d == 2) ? (ACTIVE - 2 * ROWS_PER_DESC) : ROWS_PER_DESC;
        const int kb = d * ROWS_PER_DESC;

        u32x4 g0;
        g0[0] = 0xC0000001u;                         // gather_mode|idx32|count=1
        g0[1] = lbase + (u32)(kb * ROW_BYTES);       // lds_addr (rows compacted)
        g0[2] = glo;                                 // global_addr[31:0]
        g0[3] = (ghi & 0x01FFFFFFu) | 0x80000000u;   // addr[56:32] | type=2

        u32x8 g1;
        g1[0] = 0x00020000u;                         // wg_mask=0, data_size=4B
        g1[1] = ((u32)LPA) << 16;                    // tensor_dim0 = 96 (lo16)
        g1[2] = (F & 0xFFFFu) << 16;                 // dim0 hi16=0 | tensor_dim1 lo16
        g1[3] = ((F >> 16) & 0xFFFFu)                // tensor_dim1 hi16
              | (((u32)LPA) << 16);                  // tile_dim0 = 96
        g1[4] = nv;                                  // tile_dim1 = #valid indices
        g1[5] = (u32)OUT_DIM;                        // tensor_dim0_stride = 576
        g1[6] = 0u;                                  // stride hi16 / dim1_stride (ignored)
        g1[7] = 0u;

        u32x4 g2, g3;                                // 8 x 32-bit row indices
#pragma unroll
        for (int j = 0; j < 4; ++j) {
            const int k0 = kb + j, k1 = kb + 4 + j;
            g2[j] = (k0 < ACTIVE) ? (u32)__builtin_amdgcn_readlane(myCol, k0) : 0u;
            g3[j] = (k1 < ACTIVE) ? (u32)__builtin_amdgcn_readlane(myCol, k1) : 0u;
        }

        asm volatile("tensor_load_to_lds %0, %1, %2, %3"
                     :: "s"(g0), "s"(g1), "s"(g2), "s"(g3)
                     : "memory");
    }

    // Drain the Tensor Data Mover (per-wave TENSORcnt; LDS region wave-private)
    asm volatile("s_wait_tensorcnt 0x0" ::: "memory");

    // Each lane accumulates 3 contiguous columns in x127 integer units.
    const int c0 = lane * 3;
    float a0 = 0.0f, a1 = 0.0f, a2 = 0.0f;
#pragma unroll
    for (int k = 0; k < ACTIVE; ++k) {
        const float v  = __int_as_float(
            __builtin_amdgcn_readlane(__float_as_int(myVal), k)); // SGPR broadcast
        const float* r = lds + k * LPA + c0;
        a0 = fmaf(rintf(r[0] * 127.0f), v, a0);   // fq_round(w,127) scaled by 127
        a1 = fmaf(rintf(r[1] * 127.0f), v, a1);
        a2 = fmaf(rintf(r[2] * 127.0f), v, a2);
    }

    // Epilogue in x127 units: +bias, leaky(0.125), clamp[-16,111], +16,
    // floor, clip[0,127], /127  ==  reference's clip/shift/floor-fq chain.
    float acc[3] = {a0, a1, a2};
#pragma unroll
    for (int j = 0; j < 3; ++j) {
        float x = acc[j] + rintf(bias[cb + c0 + j] * 127.0f);
        x = (x >= 0.0f) ? x : 0.125f * x;
        x = fminf(fmaxf(x, -16.0f), 111.0f) + 16.0f;
        x = floorf(x);
        x = fminf(fmaxf(x, 0.0f), 127.0f);
        __builtin_nontemporal_store(x * (1.0f / 127.0f),
                                    &out[(size_t)s * LPA + c0 + j]);
    }
}

extern "C" void kernel_launch(void* const* d_in, const int* in_sizes, int n_in,
                              void* d_out, int out_size, void* d_ws, size_t ws_size,
                              hipStream_t stream)
{
    (void)n_in; (void)out_size; (void)d_ws; (void)ws_size;

    // setup_inputs() order:
    //  0: feature_indices [2, NNZ] int32   (row 0 = batch row ids, row 1 = cols)
    //  1: values          [NNZ]    float32
    //  2: weight          [F, 576] float32
    //  3: bias            [576]    float32
    //  4: ply             [B]      int32
    //  5: batch_size (device scalar, unused)  6: in_features (device scalar, unused)
    const int    nnz    = in_sizes[1];
    const int    batch  = in_sizes[4];
    const u32    F      = (u32)(in_sizes[2] / OUT_DIM);
    const int*   fi     = (const int*)d_in[0];
    const int*   colsp  = fi + nnz;               // second row of the COO index
    const float* values = (const float*)d_in[1];
    const float* weight = (const float*)d_in[2];
    const float* bias   = (const float*)d_in[3];
    const int*   ply    = (const int*)d_in[4];
    float*       out    = (float*)d_out;

    const int blocks = (batch + WAVES_PER_BLOCK - 1) / WAVES_PER_BLOCK;
    nnue_phase_tdm_kernel<<<blocks, WAVES_PER_BLOCK * 32, 0, stream>>>(
        colsp, values, weight, bias, ply, out, batch, F);
}